// motion_blur_degradation_17970143167176
// MI455X (gfx1250) — compile-verified
//
#include <hip/hip_runtime.h>

#ifndef __has_builtin
#define __has_builtin(x) 0
#endif

namespace {
constexpr int kB = 4, kC = 3, kH = 512, kW = 512, kS = 16;
constexpr int kPlane   = kH * kW;          // 262144 elements = 1 MB
constexpr int kThreads = 256;              // 8 wave32 per block
constexpr int kPixels  = kB * kH * kW;     // 1,048,576 threads total

// 8-byte row-pair load, only 4-byte aligned (x is arbitrary).
struct F2 { float lo, hi; };

typedef __attribute__((address_space(1))) int GInt;  // printed as '__device__ int *'
typedef __attribute__((address_space(3))) int LInt;  // LDS
} // namespace

__global__ __launch_bounds__(kThreads) void motion_blur_kernel(
    const float* __restrict__ img,
    const float* __restrict__ dxp,
    float* __restrict__ out)
{
  __shared__ float s_d[2 * kThreads];

  const int tid = (int)threadIdx.x;
  const unsigned idx = blockIdx.x * (unsigned)kThreads + (unsigned)tid;
  const int x = (int)(idx & (kW - 1));          // W = 2^9
  const int y = (int)((idx >> 9) & (kH - 1));   // H = 2^9
  const int b = (int)(idx >> 18);

  // Per-pixel motion vector addresses: dx laid out (B, 2, H, W).
  const float* gdx = dxp + (size_t)(b * 2) * kPlane + (size_t)y * kW + x;
  const float* gdy = gdx + kPlane;

  // ---- CDNA5 async global->LDS staging of the motion vectors (ASYNCcnt) ----
#if __has_builtin(__builtin_amdgcn_global_load_async_to_lds_b32)
  __builtin_amdgcn_global_load_async_to_lds_b32(
      (GInt*)(uintptr_t)gdx, (LInt*)&s_d[tid], 0, 0);
  __builtin_amdgcn_global_load_async_to_lds_b32(
      (GInt*)(uintptr_t)gdy, (LInt*)&s_d[kThreads + tid], 0, 0);
#else
  {
    unsigned l0 = (unsigned)(unsigned long long)
        (__attribute__((address_space(3))) float*)&s_d[tid];
    unsigned l1 = l0 + (unsigned)(kThreads * sizeof(float));
    asm volatile("global_load_async_to_lds_b32 %0, %2, off\n\t"
                 "global_load_async_to_lds_b32 %1, %3, off"
                 :: "v"(l0), "v"(l1),
                    "v"((unsigned long long)(uintptr_t)gdx),
                    "v"((unsigned long long)(uintptr_t)gdy)
                 : "memory");
  }
#endif
#if __has_builtin(__builtin_amdgcn_s_wait_asynccnt)
  __builtin_amdgcn_s_wait_asynccnt(0);
#else
  asm volatile("s_wait_asynccnt 0" ::: "memory");
#endif
  asm volatile("" ::: "memory");  // fence: LDS now holds dx/dy

  const float dxv = s_d[tid];
  const float dyv = s_d[kThreads + tid];

  const float* imgb = img + (size_t)b * (size_t)(kC * kPlane);
  const float fx = (float)x, fy = (float)y;
  float acc0 = 0.0f, acc1 = 0.0f, acc2 = 0.0f;

#pragma unroll
  for (int s = 0; s < kS; ++s) {
    const float t = ((float)s + 0.5f) * (1.0f / (float)kS) - 0.5f;
    float px = fmaf(t, dxv, fx);
    float py = fmaf(t, dyv, fy);
    px = fminf(fmaxf(px, 0.0f), (float)(kW - 1));
    py = fminf(fmaxf(py, 0.0f), (float)(kH - 1));
    const float x0f = floorf(px);
    const float y0f = floorf(py);
    float wx = px - x0f;
    float wy = py - y0f;
    int x0 = (int)x0f;
    int y0 = (int)y0f;
    // Border fold: x0==W-1 implies wx==0 exactly (px clamped to W-1), so
    // shifting the 2-wide window left and setting wx=1 selects img[W-1].
    if (x0 > kW - 2) { x0 = kW - 2; wx = 1.0f; }
    if (y0 > kH - 2) { y0 = kH - 2; wy = 1.0f; }

    const float* r0 = imgb + (size_t)y0 * kW + x0;   // row y0, cols [x0, x0+1]
    const float* r1 = r0 + kW;                       // row y0+1

    // 6 x 8-byte gathers cover all 4 corners of all 3 channels.
    // Channel planes are 1 MB apart -> fits signed 24-bit IOFFSET (saddr form).
    const F2 a0 = *(const F2*)(r0);
    const F2 b0 = *(const F2*)(r1);
    const F2 a1 = *(const F2*)(r0 + kPlane);
    const F2 b1 = *(const F2*)(r1 + kPlane);
    const F2 a2 = *(const F2*)(r0 + 2 * kPlane);
    const F2 b2 = *(const F2*)(r1 + 2 * kPlane);

    {
      const float top = fmaf(wx, a0.hi - a0.lo, a0.lo);
      const float bot = fmaf(wx, b0.hi - b0.lo, b0.lo);
      acc0 += fmaf(wy, bot - top, top);
    }
    {
      const float top = fmaf(wx, a1.hi - a1.lo, a1.lo);
      const float bot = fmaf(wx, b1.hi - b1.lo, b1.lo);
      acc1 += fmaf(wy, bot - top, top);
    }
    {
      const float top = fmaf(wx, a2.hi - a2.lo, a2.lo);
      const float bot = fmaf(wx, b2.hi - b2.lo, b2.lo);
      acc2 += fmaf(wy, bot - top, top);
    }
  }

  const size_t o = (size_t)b * (size_t)(kC * kPlane) + (size_t)y * kW + x;
  out[o]              = acc0 * (1.0f / (float)kS);
  out[o + kPlane]     = acc1 * (1.0f / (float)kS);
  out[o + 2 * kPlane] = acc2 * (1.0f / (float)kS);
}

extern "C" void kernel_launch(void* const* d_in, const int* in_sizes, int n_in,
                              void* d_out, int out_size, void* d_ws, size_t ws_size,
                              hipStream_t stream) {
  (void)in_sizes; (void)n_in; (void)out_size; (void)d_ws; (void)ws_size;
  const float* img = (const float*)d_in[0];   // (4,3,512,512) f32
  const float* dx  = (const float*)d_in[1];   // (4,2,512,512) f32
  float*       out = (float*)d_out;           // (4,3,512,512) f32

  const dim3 grid(kPixels / kThreads);        // 4096 blocks
  const dim3 block(kThreads);                 // 256 threads = 8 waves (wave32)
  hipLaunchKernelGGL(motion_blur_kernel, grid, block, 0, stream, img, dx, out);
}